// DiffVorticeSketchRender_35210141892890
// MI455X (gfx1250) — compile-verified
//
#include <hip/hip_runtime.h>
#include <math.h>

// -------------------------------------------------------------------------
// DiffVorticeSketchRender for MI455X (gfx1250, wave32).
// Memory-bound pipeline (~0.9 GB traffic @ 23.3 TB/s ≈ 40 us). Separable
// 7-tap Gaussian passes are banded-Toeplitz matmuls on the f32 WMMA pipe:
// D(16x16) = A(16x24 band) x B(24x16) via 6x V_WMMA_F32_16X16X4_F32 with
// exact f32 FMA precision. Strides are compile-time so B-pair loads widen
// to b64 and x-pass stores widen to b128; interior conv tiles (wave-uniform
// ct in [1,6]) take a mask-free load path.
// Workspace requirement: 2 ping-pong buffers of 8 volumes = 128 MiB.
// -------------------------------------------------------------------------

typedef __attribute__((ext_vector_type(2))) float v2f;
typedef __attribute__((ext_vector_type(8))) float v8f;

#define VOL (128 * 128 * 128)   // 2097152 elements per volume

// ---- Kernel 1: vorticity magnitude |curl(v)| ----------------------------
// fdiff(f)[i] = f[min(i,n-2)+1] - f[min(i,n-2)]  (forward diff, last repeated)
__global__ void __launch_bounds__(256)
curlmag_kernel(const float* __restrict__ v, float* __restrict__ vn)
{
    int idx = blockIdx.x * blockDim.x + threadIdx.x;   // 0 .. 4*VOL-1
    int b = idx >> 21;
    int r = idx & (VOL - 1);
    int z = r >> 14;
    int y = (r >> 7) & 127;
    int x = r & 127;

    const float* U  = v + (size_t)b * 3 * VOL;  // v[:,0]
    const float* Vv = U + VOL;                  // v[:,1]
    const float* W  = Vv + VOL;                 // v[:,2]

    int z0 = z < 126 ? z : 126;
    int y0 = y < 126 ? y : 126;
    int x0 = x < 126 ? x : 126;
    int bz = z0 * 16384 + y * 128 + x;   // base for d/dz
    int by = z * 16384 + y0 * 128 + x;   // base for d/dy
    int bx = z * 16384 + y * 128 + x0;   // base for d/dx

    float dWdy = W[by + 128]    - W[by];
    float dVdz = Vv[bz + 16384] - Vv[bz];
    float dUdz = U[bz + 16384]  - U[bz];
    float dWdx = W[bx + 1]      - W[bx];
    float dVdx = Vv[bx + 1]     - Vv[bx];
    float dUdy = U[by + 128]    - U[by];

    float cu = dWdy - dVdz;   // fdiff(w,H) - fdiff(v,D)
    float cv = dUdz - dWdx;   // fdiff(u,D) - fdiff(w,W)
    float cw = dVdx - dUdy;   // fdiff(v,W) - fdiff(u,H)

    vn[(size_t)b * VOL + r] = sqrtf(cu * cu + cv * cv + cw * cw);
}

// ---- Banded Toeplitz coefficient, branchless, exploits w[d] == w[6-d] ----
// Returns w[d] for d in [0,6], 0 otherwise (covers out-of-band + k>=22 pad).
__device__ __forceinline__ float bandw(int d, float w0, float w1, float w2, float w3)
{
    int e = abs(d - 3);
    float r = 0.0f;
    r = (e == 3) ? w0 : r;
    r = (e == 2) ? w1 : r;
    r = (e == 1) ? w2 : r;
    r = (e == 0) ? w3 : r;
    return r;
}

// ---- Kernel 2/3/4: generic 1-D 7-tap conv pass via f32 WMMA -------------
// One wave = one tile of 16 conv-axis outputs x 16 lines.
// out[m] = sum_k A[m,k] * in[p0+k],  A[m,k] = w[k-m] for 0<=k-m<=6, K=0..23.
// Zero padding at conv-axis boundaries (matches conv_general_dilated).
// SC/SN/SO: element strides of conv axis / line axis / remaining axis.
template <int SC, int SN, int SO>
__global__ void __launch_bounds__(256)
conv_pass_kernel(const float* __restrict__ src0,   // volumes 0..3 (d channel)
                 const float* __restrict__ src1,   // volumes 4..7 (vn channel)
                 float* __restrict__ dst,          // 8 volumes
                 float w0, float w1, float w2, float w3)
{
    int lane = threadIdx.x & 31;
    int ct   = threadIdx.x >> 5;          // wave id -> conv-axis tile (0..7)
    int h    = lane >> 4;                 // half-wave
    int n    = lane & 15;                 // B/D column (line index in tile)
    int m    = n;                         // A row

    int nt  = blockIdx.x & 7;             // line-axis tile (0..7)
    int o   = (blockIdx.x >> 3) & 127;    // "other" axis coordinate
    int vol = blockIdx.x >> 10;           // volume (0..7)

    const float* src = (vol < 4) ? (src0 + (size_t)vol * VOL)
                                 : (src1 + (size_t)(vol - 4) * VOL);
    float* out = dst + (size_t)vol * VOL;

    int lineOff = o * SO + (nt * 16 + n) * SN;
    int p0 = ct * 16 - 3;                 // first input row of the K=24 band

    // A matrix per-lane values (f32 16x4 layout: VGPR0<->K=2h, VGPR1<->K=2h+1)
    v2f Areg[6];
#pragma unroll
    for (int c = 0; c < 6; ++c) {
        int k0 = 4 * c + 2 * h;
        Areg[c][0] = bandw(k0 - m,     w0, w1, w2, w3);
        Areg[c][1] = bandw(k0 + 1 - m, w0, w1, w2, w3);
    }

    v2f Breg[6];
    if (ct >= 1 && ct <= 6) {
        // Interior tile (wave-uniform branch): band rows p0..p0+21 all lie in
        // [13,116] and rows 22/23 (still in range) hit zero A columns, so load
        // all 24 rows unconditionally — constant immediate offsets off one base.
        const float* bp = src + lineOff + (p0 + 2 * h) * SC;
#pragma unroll
        for (int c = 0; c < 6; ++c) {
            Breg[c][0] = bp[(4 * c) * SC];       // SC==1: pair merges to b64
            Breg[c][1] = bp[(4 * c + 1) * SC];
        }
    } else {
        // Edge tile: branchless clamped loads + cndmask zero padding.
#pragma unroll
        for (int c = 0; c < 6; ++c) {
            int k0 = 4 * c + 2 * h;
            int pa = p0 + k0;
            int pb = pa + 1;
            int pac = pa < 0 ? 0 : (pa > 127 ? 127 : pa);   // always-valid addr
            int pbc = pb < 0 ? 0 : (pb > 127 ? 127 : pb);
            float ra = src[lineOff + pac * SC];
            float rb = src[lineOff + pbc * SC];
            Breg[c][0] = (pa == pac) ? ra : 0.0f;
            Breg[c][1] = (pb == pbc) ? rb : 0.0f;
        }
    }

    v8f acc = {0.f, 0.f, 0.f, 0.f, 0.f, 0.f, 0.f, 0.f};
#pragma unroll
    for (int c = 0; c < 6; ++c) {
        // 8 args: (neg_a, A, neg_b, B, c_mod, C, reuse_a, reuse_b)
        acc = __builtin_amdgcn_wmma_f32_16x16x4_f32(
            false, Areg[c], false, Breg[c], (short)0, acc, false, false);
    }

    // C/D layout: VGPR r holds M = r (lanes 0-15) / M = r+8 (lanes 16-31).
    // Immediate offsets off one base; SC==1: 8 consecutive floats -> b128 x2.
    float* op = out + lineOff + (ct * 16 + 8 * h) * SC;
#pragma unroll
    for (int r2 = 0; r2 < 8; ++r2) {
        op[r2 * SC] = acc[r2];
    }
}

// ---- Kernel 5: flip + cumsum + transmittance + trapezoid integral -------
__global__ void __launch_bounds__(256)
integrate_kernel(const float* __restrict__ dsbuf,  // smoothed d, vols 0..3
                 const float* __restrict__ vnbuf,  // smoothed vn, vols 0..3
                 float* __restrict__ outp)
{
    int tid = blockIdx.x * blockDim.x + threadIdx.x;  // 0..65535
    int b  = tid >> 14;
    int yx = tid & 16383;
    const float* ds = dsbuf + (size_t)b * VOL + yx;
    const float* vn = vnbuf + (size_t)b * VOL + yx;

    const float Cc = 20.0f;

    // k = 0 corresponds to z = 127 (axis-2 flip)
    float xacc = ds[127 * 16384];
    float t    = (Cc * xacc + 1.0f) * expf(-Cc * xacc);
    float vf   = vn[127 * 16384];
    float front = (1.0f - t) * vf;
    float trap  = 0.0f;

    for (int k = 1; k < 128; ++k) {
        int z = 127 - k;
        xacc += ds[z * 16384];
        float tn  = (Cc * xacc + 1.0f) * expf(-Cc * xacc);
        float vfn = vn[z * 16384];
        trap += (t - tn) * (vf + vfn) * 0.5f;
        t = tn;
        vf = vfn;
    }

    float res = front + trap;
    outp[tid] = fminf(fmaxf(res, 0.0f), 1.0f);
}

// -------------------------------------------------------------------------
extern "C" void kernel_launch(void* const* d_in, const int* in_sizes, int n_in,
                              void* d_out, int out_size, void* d_ws, size_t ws_size,
                              hipStream_t stream)
{
    const float* d = (const float*)d_in[0];   // (4,1,128,128,128)
    const float* v = (const float*)d_in[1];   // (4,3,128,128,128)
    float* out = (float*)d_out;               // (4,1,128,128) = 65536 floats

    // Workspace: two ping-pong buffers of 8 volumes each (128 MiB total).
    float* buf0 = (float*)d_ws;                    // 8 * VOL floats
    float* buf1 = buf0 + (size_t)8 * VOL;          // 8 * VOL floats
    float* vnraw = buf0 + (size_t)4 * VOL;         // vols 4..7 of buf0

    // Normalized separable 1-D Gaussian (sigma = 1.6), host-side in double.
    // Symmetric: only w[0..3] needed (w[d] == w[6-d]).
    float w0, w1, w2, w3;
    {
        double gv[7], s = 0.0;
        for (int i = 0; i < 7; ++i) {
            double x = (double)(i - 3) / 1.6;
            gv[i] = exp(-x * x / 2.0);
            s += gv[i];
        }
        w0 = (float)(gv[0] / s);
        w1 = (float)(gv[1] / s);
        w2 = (float)(gv[2] / s);
        w3 = (float)(gv[3] / s);
    }

    // 1) |curl(v)| -> buf0 vols 4..7
    curlmag_kernel<<<(4 * VOL) / 256, 256, 0, stream>>>(v, vnraw);

    // 2) x-pass (conv stride 1, lines along y, other = z): d & vnraw -> buf1
    conv_pass_kernel<1, 128, 16384><<<8192, 256, 0, stream>>>(
        d, vnraw, buf1, w0, w1, w2, w3);
    // 3) y-pass (conv stride 128, lines along x, other = z): buf1 -> buf0
    conv_pass_kernel<128, 1, 16384><<<8192, 256, 0, stream>>>(
        buf1, buf1 + (size_t)4 * VOL, buf0, w0, w1, w2, w3);
    // 4) z-pass (conv stride 16384, lines along x, other = y): buf0 -> buf1
    conv_pass_kernel<16384, 1, 128><<<8192, 256, 0, stream>>>(
        buf0, buf0 + (size_t)4 * VOL, buf1, w0, w1, w2, w3);

    // 5) depth integration -> out
    integrate_kernel<<<65536 / 256, 256, 0, stream>>>(buf1, buf1 + (size_t)4 * VOL, out);
}